// GCN_39573828665766
// MI455X (gfx1250) — compile-verified
//
#include <hip/hip_runtime.h>
#include <hip/hip_bf16.h>

#define NN 50000
#define NE 800000
#define D  128
#define NG 512
#define NCLS 10
#define BN_EPS 1e-5f

typedef __attribute__((ext_vector_type(2))) float v2f;
typedef __attribute__((ext_vector_type(8))) float v8f;

// ---------------- degree / dinv ----------------
__global__ void k_fill_deg(float* deg) {
    int i = blockIdx.x * 256 + threadIdx.x;
    if (i < NN) deg[i] = 1.0f;               // self-loop
}
__global__ void k_count_deg(const int* __restrict__ dst, float* deg) {
    int e = blockIdx.x * 256 + threadIdx.x;
    if (e < NE) atomicAdd(&deg[dst[e]], 1.0f);
}
__global__ void k_dinv(float* deg) {
    int i = blockIdx.x * 256 + threadIdx.x;
    if (i < NN) deg[i] = rsqrtf(deg[i]);
}

// ---------------- WMMA f32 GEMM: Y[NN x 128] = H[NN x 128] * W[128 x 128] ----------------
// grid 625 blocks x 256 threads (8 waves). Each block: 80 rows; wave w owns n-tile w,
// loops over 5 m-tiles. W staged in LDS (64 KB).
__global__ void __launch_bounds__(256) k_gemm_wmma(const float* __restrict__ H,
                                                   const float* __restrict__ W,
                                                   float* __restrict__ Y) {
    __shared__ float sW[D * D];
    int tid = threadIdx.x;
    for (int i = tid; i < D * D / 4; i += 256)
        ((float4*)sW)[i] = ((const float4*)W)[i];
    __syncthreads();

    int lane   = tid & 31;
    int nbase  = (tid >> 5) << 4;       // wave id * 16
    int col    = lane & 15;             // N within tile / M row within A
    int khalf  = (lane >> 4) << 1;      // 0 or 2  (A/B K sub-offset per ISA layout)
    int moff   = (lane >> 4) << 3;      // 0 or 8  (C/D row offset per ISA layout)

    for (int mt = 0; mt < 5; mt++) {
        int mbase = blockIdx.x * 80 + mt * 16;
        const float* hrow = H + (size_t)(mbase + col) * D;
        v8f c = {};
        #pragma unroll
        for (int k = 0; k < D; k += 4) {
            v2f a, b;
            a[0] = hrow[k + khalf];
            a[1] = hrow[k + khalf + 1];
            b[0] = sW[(k + khalf) * D + nbase + col];
            b[1] = sW[(k + khalf + 1) * D + nbase + col];
            c = __builtin_amdgcn_wmma_f32_16x16x4_f32(false, a, false, b,
                                                      (short)0, c, false, false);
        }
        float* yp = Y + (size_t)(mbase + moff) * D + nbase + col;
        #pragma unroll
        for (int r = 0; r < 8; r++)
            yp[(size_t)r * D] = c[r];
    }
}

// ---------------- agg init: agg = hW * dinv[i]^2 + bias[f] ----------------
__global__ void k_init_agg(const float* __restrict__ hW, const float* __restrict__ dinv,
                           const float* __restrict__ bias, float* __restrict__ agg) {
    size_t idx = (size_t)blockIdx.x * 256 + threadIdx.x;
    if (idx >= (size_t)NN * D) return;
    int i = (int)(idx >> 7);
    int f = (int)(idx & (D - 1));
    float dv = dinv[i];
    agg[idx] = hW[idx] * dv * dv + bias[f];
}

// ---------------- edge scatter: agg[dst] += hW[src] * dinv[s]*dinv[d] ----------------
// one wave32 per edge, 4 features per lane
__global__ void k_edge_scatter(const int* __restrict__ src, const int* __restrict__ dst,
                               const float* __restrict__ dinv,
                               const float* __restrict__ hW, float* __restrict__ agg) {
    int e = blockIdx.x * 8 + (threadIdx.x >> 5);
    if (e >= NE) return;
    int lane = threadIdx.x & 31;
    int s = src[e], d = dst[e];
    float norm = dinv[s] * dinv[d];
    float4 v = ((const float4*)(hW + (size_t)s * D))[lane];
    float* ap = agg + (size_t)d * D + lane * 4;
    atomicAdd(ap + 0, v.x * norm);
    atomicAdd(ap + 1, v.y * norm);
    atomicAdd(ap + 2, v.z * norm);
    atomicAdd(ap + 3, v.w * norm);
}

// ---------------- batchnorm stats ----------------
__global__ void k_zero_stats(float* gsum, float* gsqs) {
    int f = threadIdx.x;
    if (f < D) { gsum[f] = 0.0f; gsqs[f] = 0.0f; }
}
__global__ void k_stats(const float* __restrict__ agg, float* gsum, float* gsqs) {
    int f = threadIdx.x;                 // 128 threads, one feature each
    int r0 = blockIdx.x * 256;
    int r1 = r0 + 256; if (r1 > NN) r1 = NN;
    float s = 0.0f, s2 = 0.0f;
    for (int r = r0; r < r1; r++) {
        float v = agg[(size_t)r * D + f];
        s += v; s2 += v * v;
    }
    atomicAdd(&gsum[f], s);
    atomicAdd(&gsqs[f], s2);
}
__global__ void k_bn_finalize(const float* gsum, const float* gsqs,
                              const float* __restrict__ gamma, const float* __restrict__ beta,
                              float* scale, float* shift) {
    int f = threadIdx.x;
    const float invn = 1.0f / (float)NN;
    float mu  = gsum[f] * invn;
    float var = gsqs[f] * invn - mu * mu;
    float sc  = gamma[f] * rsqrtf(var + BN_EPS);
    scale[f] = sc;
    shift[f] = beta[f] - mu * sc;
}
__global__ void k_bn_apply_relu(const float* __restrict__ agg, const float* __restrict__ scale,
                                const float* __restrict__ shift, float* __restrict__ out) {
    size_t idx = (size_t)blockIdx.x * 256 + threadIdx.x;
    if (idx >= (size_t)NN * D) return;
    int f = (int)(idx & (D - 1));
    float v = agg[idx] * scale[f] + shift[f];
    out[idx] = fmaxf(v, 0.0f);
}

// ---------------- global mean pool + final linear ----------------
__global__ void k_zero_pool(float* pooled, float* counts) {
    int i = blockIdx.x * 256 + threadIdx.x;
    if (i < NG * D) pooled[i] = 0.0f;
    if (i < NG) counts[i] = 0.0f;
}
__global__ void k_pool_scatter(const int* __restrict__ batch, const float* __restrict__ h,
                               float* __restrict__ pooled, float* __restrict__ counts) {
    int i = blockIdx.x * 8 + (threadIdx.x >> 5);
    if (i >= NN) return;
    int lane = threadIdx.x & 31;
    int g = batch[i];
    float4 v = ((const float4*)(h + (size_t)i * D))[lane];
    float* pp = pooled + (size_t)g * D + lane * 4;
    atomicAdd(pp + 0, v.x);
    atomicAdd(pp + 1, v.y);
    atomicAdd(pp + 2, v.z);
    atomicAdd(pp + 3, v.w);
    if (lane == 0) atomicAdd(&counts[g], 1.0f);
}
__global__ void k_final_linear(const float* __restrict__ pooled, const float* __restrict__ counts,
                               const float* __restrict__ Wl, const float* __restrict__ bl,
                               float* __restrict__ out) {
    int g = blockIdx.x;
    int c = threadIdx.x;
    if (c >= NCLS) return;
    float inv = 1.0f / fmaxf(counts[g], 1.0f);
    float acc = bl[c];
    for (int f = 0; f < D; f++)
        acc += pooled[(size_t)g * D + f] * inv * Wl[f * NCLS + c];
    out[g * NCLS + c] = acc;
}

extern "C" void kernel_launch(void* const* d_in, const int* in_sizes, int n_in,
                              void* d_out, int out_size, void* d_ws, size_t ws_size,
                              hipStream_t stream) {
    (void)in_sizes; (void)n_in; (void)out_size; (void)ws_size;

    const float* x     = (const float*)d_in[0];
    const int*   edges = (const int*)d_in[1];
    const int*   batch = (const int*)d_in[2];
    const float* W1    = (const float*)d_in[3];
    const float* b1    = (const float*)d_in[4];
    const float* g1    = (const float*)d_in[5];
    const float* be1   = (const float*)d_in[6];
    const float* Wc    = (const float*)d_in[7];
    const float* bc    = (const float*)d_in[8];
    const float* gc    = (const float*)d_in[9];
    const float* bec   = (const float*)d_in[10];
    const float* Wl    = (const float*)d_in[11];
    const float* bl    = (const float*)d_in[12];
    float* out = (float*)d_out;

    const int* src = edges;
    const int* dst = edges + NE;

    // workspace layout (floats)
    float* hA     = (float*)d_ws;            // node features (layer input), NN*D
    float* hB     = hA + (size_t)NN * D;     // gemm output, NN*D
    float* hC     = hB + (size_t)NN * D;     // aggregate,   NN*D
    float* dinv   = hC + (size_t)NN * D;     // NN
    float* gsum   = dinv + NN;               // D
    float* gsqs   = gsum + D;                // D
    float* scale  = gsqs + D;                // D
    float* shift  = scale + D;               // D
    float* pooled = shift + D;               // NG*D
    float* counts = pooled + (size_t)NG * D; // NG

    const int elemBlocks = (NN * D + 255) / 256;        // 25000
    const int edgeBlocks = (NE + 7) / 8;                // 100000
    const int nodeBlocks8 = (NN + 7) / 8;               // 6250
    const int statBlocks = (NN + 255) / 256;            // 196

    // degrees -> dinv
    k_fill_deg <<<(NN + 255) / 256, 256, 0, stream>>>(dinv);
    k_count_deg<<<(NE + 255) / 256, 256, 0, stream>>>(dst, dinv);
    k_dinv     <<<(NN + 255) / 256, 256, 0, stream>>>(dinv);

    for (int l = 0; l < 4; l++) {
        const float* H     = (l == 0) ? x  : hA;
        const float* W     = (l == 0) ? W1 : Wc  + (size_t)(l - 1) * D * D;
        const float* bias  = (l == 0) ? b1 : bc  + (size_t)(l - 1) * D;
        const float* gamma = (l == 0) ? g1 : gc  + (size_t)(l - 1) * D;
        const float* beta  = (l == 0) ? be1 : bec + (size_t)(l - 1) * D;

        k_gemm_wmma    <<<625, 256, 0, stream>>>(H, W, hB);
        k_zero_stats   <<<1, 128, 0, stream>>>(gsum, gsqs);
        k_init_agg     <<<elemBlocks, 256, 0, stream>>>(hB, dinv, bias, hC);
        k_edge_scatter <<<edgeBlocks, 256, 0, stream>>>(src, dst, dinv, hB, hC);
        k_stats        <<<statBlocks, 128, 0, stream>>>(hC, gsum, gsqs);
        k_bn_finalize  <<<1, 128, 0, stream>>>(gsum, gsqs, gamma, beta, scale, shift);
        k_bn_apply_relu<<<elemBlocks, 256, 0, stream>>>(hC, scale, shift, hA);
    }

    k_zero_pool    <<<(NG * D + 255) / 256, 256, 0, stream>>>(pooled, counts);
    k_pool_scatter <<<nodeBlocks8, 256, 0, stream>>>(batch, hA, pooled, counts);
    k_final_linear <<<NG, 32, 0, stream>>>(pooled, counts, Wl, bl, out);
}